// Net_27882927686183
// MI455X (gfx1250) — compile-verified
//
#include <hip/hip_runtime.h>
#include <math.h>

// CDNA5 wave32 WMMA vector types
typedef __attribute__((ext_vector_type(2))) float v2f;   // A/B for 16x16x4 f32 (2 VGPRs)
typedef __attribute__((ext_vector_type(8))) float v8f;   // C/D 16x16 f32 (8 VGPRs)
typedef __attribute__((ext_vector_type(2))) int   v2i;

#define N_IN   100
#define N_HID  501
#define N_TOT  601
#define MAXLVL 512   // level values are in [1, 501]

// Fast sigmoid: one v_exp_f32 + one v_rcp_f32 on the serial critical path.
__device__ __forceinline__ float sigmoid_fast(float t) {
    return __builtin_amdgcn_rcpf(1.0f + __expf(-t));
}

// Select element r (0..7): 3-deep selection tree.
__device__ __forceinline__ float pick8(v8f c, int r) {
    const float t0 = (r & 1) ? c[1] : c[0];
    const float t1 = (r & 1) ? c[3] : c[2];
    const float t2 = (r & 1) ? c[5] : c[4];
    const float t3 = (r & 1) ? c[7] : c[6];
    const float u0 = (r & 2) ? t1 : t0;
    const float u1 = (r & 2) ? t3 : t2;
    return (r & 4) ? u1 : u0;
}

__global__ __launch_bounds__(32)
void dag_net_kernel(const float* __restrict__ x,
                    const float* __restrict__ w_in,
                    const float* __restrict__ b_in,
                    const float* __restrict__ w_hid,   // [501,4]
                    const float* __restrict__ b_hid,   // [501]
                    const int*   __restrict__ edges,   // [501,4]
                    float* __restrict__ out)           // [1]
{
    // Unified value array: [0,100) inputs | [100,601) SLOT-ordered hidden |
    // [601,617) per-lane dummy slots. Writer destination = N_IN + slot (pure
    // ALU, no dstn[] load after the WMMA).
    __shared__ float uval[N_IN + N_HID + 16];
    __shared__ int   lev[N_TOT];
    __shared__ int   elds[N_HID * 4];   // edge indices (node order, for levelization)
    __shared__ float wslot[N_HID * 4];  // weights permuted into slot order
    __shared__ int   eslot[N_HID * 4];  // edge BYTE offsets into uval, slot order
    __shared__ float bslot[N_HID];      // biases permuted into slot order
    __shared__ int   slotOf[N_HID];     // hidden node -> slot (inverse of order)
    __shared__ int   cnt[MAXLVL];       // per-level node counts
    __shared__ int   lvl2[MAXLVL * 2];  // packed {start,end} per level
    __shared__ int   cur[MAXLVL];       // scatter cursors
    __shared__ int   order[N_HID];      // hidden node ids grouped by level
    __shared__ int   meta;              // max level
    __shared__ int   changed;           // relaxation convergence flag

    const int lane = threadIdx.x;

    __builtin_prefetch(w_hid, 0, 0);
    __builtin_prefetch(edges, 0, 0);

    // ---- Phase 0: stage edges to LDS, input layer, init levels ----
    for (int i = lane; i < N_HID * 4; i += 32) elds[i] = edges[i];
    for (int i = lane; i < N_HID; i += 32)     lev[N_IN + i] = 1;
    for (int i = lane; i < N_IN; i += 32) {
        uval[i] = sigmoid_fast(fmaf(w_in[i], x[i], b_in[i]));
        lev[i]  = 0;
    }
    for (int l = lane; l < MAXLVL; l += 32) cnt[l] = 0;
    if (lane == 0) meta = 0;
    __syncthreads();

    // ---- Phase 1: parallel chaotic relaxation to exact levels ----
    for (;;) {
        if (lane == 0) changed = 0;
        __syncthreads();
        int loc = 0;
        for (int k = lane; k < N_HID; k += 32) {
            int m = 0;
#pragma unroll
            for (int j = 0; j < 4; ++j) {
                const int l = lev[elds[k * 4 + j]];
                m = (l > m) ? l : m;
            }
            if (m + 1 != lev[N_IN + k]) { lev[N_IN + k] = m + 1; loc = 1; }
        }
        if (loc) changed = 1;           // benign same-value race
        __syncthreads();
        if (!changed) break;            // uniform LDS read
    }

    // ---- Phase 2: histogram + max level (LDS atomics) ----
    {
        int mloc = 0;
        for (int k = lane; k < N_HID; k += 32) {
            const int l = lev[N_IN + k];
            atomicAdd(&cnt[l], 1);
            mloc = (l > mloc) ? l : mloc;
        }
        atomicMax(&meta, mloc);
    }
    __syncthreads();
    const int maxL = meta;

    // ---- Phase 3: prefix scan -> packed {start,end} per level ----
    {
        int carry = 0;
        const int nch = (maxL + 32) / 32;
        for (int c = 0; c < nch; ++c) {
            const int l    = c * 32 + lane;
            const int orig = (l >= 1 && l <= maxL) ? cnt[l] : 0;
            int v = orig;
#pragma unroll
            for (int off = 1; off < 32; off <<= 1) {
                const int t = __shfl_up(v, off, 32);
                if (lane >= off) v += t;
            }
            if (l >= 1 && l <= maxL) {
                const int st = carry + (v - orig);
                lvl2[2 * l]     = st;          // start
                lvl2[2 * l + 1] = carry + v;   // end (= start + cnt)
                cur[l]          = st;
            }
            carry += __shfl(v, 31, 32);
        }
    }
    __syncthreads();

    // ---- Phase 4: scatter node ids by level + inverse map node->slot ----
    for (int k = lane; k < N_HID; k += 32) {
        const int pos = atomicAdd(&cur[lev[N_IN + k]], 1);
        order[pos]  = N_IN + k;
        slotOf[k]   = pos;
    }
    __syncthreads();

    // ---- Phase 4.5: permute tables into slot order; edge targets translated
    //      into uval byte offsets (inputs keep id, hidden -> N_IN + slot) ----
    for (int slot = lane; slot < N_HID; slot += 32) {
        const int node = order[slot];
        const int hid  = node - N_IN;
#pragma unroll
        for (int j = 0; j < 4; ++j) {
            wslot[slot * 4 + j] = w_hid[hid * 4 + j];      // global, L2-resident
            const int ei = elds[hid * 4 + j];
            const int ui = (ei < N_IN) ? ei : (N_IN + slotOf[ei - N_IN]);
            eslot[slot * 4 + j] = ui << 2;                 // byte offset into uval
        }
        bslot[slot] = b_hid[hid];
    }
    __syncthreads();

    // ---- Phase 5: level-parallel evaluation, 16 nodes per WMMA ----
    // A (16x4 f32): lane L holds row M=L&15, K={0,1} for L<16 else {2,3}.
    // B (4x16 f32): lane L holds col N=L&15, same K split. D diagonal = 16 dots.
    const int nsel = lane & 15;
    const int k0   = (lane < 16) ? 0 : 2;
    const bool writer = (lane < 8) || (lane >= 24);
    const int  reg    = (lane < 8) ? lane : ((lane >= 24) ? (lane - 24) : 0);
    const int  dummy  = N_IN + N_HID + nsel;   // per-lane dummy slot

    // Level bounds in registers, striped across lanes; v_readlane broadcast.
    int sreg = 0, ereg = 0, curChunk = -1;

    for (int L = 1; L <= maxL; ++L) {
        const int c = L >> 5;
        if (c != curChunk) {               // uniform; once per 32 levels
            const v2i t = *(const v2i*)(lvl2 + 2 * (c * 32 + lane));
            sreg = t.x;
            ereg = t.y;
            curChunk = c;
        }
        const int s = __builtin_amdgcn_readlane(sreg, L & 31);
        const int e = __builtin_amdgcn_readlane(ereg, L & 31);

        for (int base = s; base < e; base += 16) {
            const int slotu = base + nsel;
            const int slot  = (slotu < e) ? slotu : (e - 1);   // clamp

            const v2f a  = *(const v2f*)(wslot + slot * 4 + k0);   // ds_load_b64
            const v2i ee = *(const v2i*)(eslot + slot * 4 + k0);   // ds_load_b64
            v2f b;
            b.x = *(const float*)((const char*)uval + ee.x);       // gather
            b.y = *(const float*)((const char*)uval + ee.y);
            const float bias = bslot[slot];

            v8f cacc = {};
            cacc = __builtin_amdgcn_wmma_f32_16x16x4_f32(
                    /*neg_a=*/false, a, /*neg_b=*/false, b,
                    /*c_mod=*/(short)0, cacc, /*reuse_a=*/false, /*reuse_b=*/false);

            const float dv = pick8(cacc, reg);
            const float nv = sigmoid_fast(dv + bias);
            // Destination is slot-indexed: pure ALU, no memory op after WMMA
            // except this store.
            const int dst = (writer && slotu < e) ? (N_IN + slotu) : dummy;
            uval[dst] = nv;
        }
        __syncthreads();   // single-wave WG: compiler/DS fence between levels
    }

    // Node 600 (the output) is hidden node 500 -> its slot-ordered location.
    if (lane == 0) out[0] = uval[N_IN + slotOf[N_HID - 1]];
}

extern "C" void kernel_launch(void* const* d_in, const int* in_sizes, int n_in,
                              void* d_out, int out_size, void* d_ws, size_t ws_size,
                              hipStream_t stream) {
    (void)in_sizes; (void)n_in; (void)out_size; (void)d_ws; (void)ws_size;
    const float* x     = (const float*)d_in[0];
    const float* w_in  = (const float*)d_in[1];
    const float* b_in  = (const float*)d_in[2];
    const float* w_hid = (const float*)d_in[3];
    const float* b_hid = (const float*)d_in[4];
    const int*   edges = (const int*)d_in[5];
    float*       out   = (float*)d_out;

    // Latency-bound sequential DAG: one workgroup, one wave, LDS-resident,
    // level-parallel with 16-wide WMMA batches per level.
    dag_net_kernel<<<1, 32, 0, stream>>>(x, w_in, b_in, w_hid, b_hid, edges, out);
}